// MultiHeadEMA_1821066133754
// MI455X (gfx1250) — compile-verified
//
#include <hip/hip_runtime.h>
#include <hip/hip_bf16.h>

typedef __attribute__((ext_vector_type(2))) float v2f;
typedef __attribute__((ext_vector_type(8))) float v8f;
typedef int v4i_t __attribute__((__vector_size__(16)));   // matches builtin proto

#define GLOBAL_AS __attribute__((address_space(1)))
#define LDS_AS    __attribute__((address_space(3)))

#if defined(__has_builtin)
#if __has_builtin(__builtin_amdgcn_global_load_async_to_lds_b128) && \
    __has_builtin(__builtin_amdgcn_s_wait_asynccnt)
#define EMA_USE_ASYNC 1
#endif
#endif
#ifndef EMA_USE_ASYNC
#define EMA_USE_ASYNC 0
#endif

constexpr int L = 4096, B = 4, D = 1024, N = 16;
constexpr int KD = 2 * D;
constexpr int CF_STRIDE = 784;                 // 256 T + 256 Wm + 256 A + 16 qC
constexpr size_t CF_BYTES = (size_t)KD * CF_STRIDE * sizeof(float);

// ---------------------------------------------------------------------------
// Kernel 1: per-channel coefficient matrices for the chunked scan.
//   T [t][s] = sum_n w_n q_n^(t-s)  (s<=t)     intra-chunk Toeplitz
//   Wm[t][n] = w_n q_n^(t+1)                   entering-state -> output
//   A [n][s] = q_n^(15-s)                      input -> end-of-chunk state
//   qC[n]    = q_n^16                          chunk-to-chunk decay
// ---------------------------------------------------------------------------
__global__ void ema_coef_kernel(const float* __restrict__ delta,
                                const float* __restrict__ alpha,
                                const float* __restrict__ beta,
                                const float* __restrict__ gamma,
                                float* __restrict__ cf) {
  const int dd  = blockIdx.x;       // 0..KD-1
  const int tid = threadIdx.x;      // 0..255
  __shared__ float w[N];
  __shared__ float qp[N][17];
  if (tid < N) {
    const int n = tid;
    const float dlt = delta[dd * N + n];
    const float alf = alpha[dd * N + n];
    const float p  = 1.0f / (1.0f + __expf(-dlt));
    const float sa = 1.0f / (1.0f + __expf(-alf));
    const float q  = 1.0f - p * sa;
    w[n] = gamma[dd * N + n] * 0.25f /* 1/sqrt(16) */ * p * beta[dd * N + n];
    float e = 1.0f;
    for (int k = 0; k <= 16; ++k) { qp[n][k] = e; e *= q; }
  }
  __syncthreads();
  float* base = cf + (size_t)dd * CF_STRIDE;
  const int t = tid >> 4, s = tid & 15;
  float Tv = 0.0f;
  if (s <= t) {
    const int lag = t - s;
#pragma unroll
    for (int n = 0; n < N; ++n) Tv += w[n] * qp[n][lag];
  }
  base[tid]       = Tv;                    // T
  base[256 + tid] = w[s] * qp[s][t + 1];   // Wm (cols = n)
  base[512 + tid] = qp[t][15 - s];         // A  (rows = n)
  if (tid < N) base[768 + tid] = qp[tid][16];
}

// ---------------------------------------------------------------------------
// Kernel 2: x (L,B,D) -> xT (B,D,L), 32x32 tiled, fully coalesced.
// ---------------------------------------------------------------------------
__global__ void ema_transpose_kernel(const float* __restrict__ x,
                                     float* __restrict__ xT) {
  __shared__ float tile[32][33];
  const int t0 = blockIdx.x * 32;
  const int d0 = blockIdx.y * 32;
  const int b  = blockIdx.z;
  for (int j = threadIdx.y; j < 32; j += 8)
    tile[j][threadIdx.x] = x[(size_t)(t0 + j) * (B * D) + b * D + d0 + threadIdx.x];
  __syncthreads();
  for (int j = threadIdx.y; j < 32; j += 8)
    xT[(size_t)(b * D + d0 + j) * L + t0 + threadIdx.x] = tile[threadIdx.x][j];
}

// ---------------------------------------------------------------------------
// Kernel 3: bidirectional EMA scan, one wave32 per (b,d) series.
// 16 superchunks x (16 chunks x 16 steps); 12x V_WMMA_F32_16X16X4_F32 per
// superchunk per direction; x staged via double-buffered
// GLOBAL_LOAD_ASYNC_TO_LDS_B128 (ASYNCcnt) when available.
// ---------------------------------------------------------------------------
__device__ __forceinline__ v8f wmma4(v2f a, v2f b, v8f c) {
  return __builtin_amdgcn_wmma_f32_16x16x4_f32(
      false, a, false, b, (short)0, c, false, false);
}

// Stage 256 contiguous floats (1 KB) from `g` into LDS buffer `buf`.
__device__ __forceinline__ void stage_superchunk(const float* g, float* buf,
                                                 int lane) {
#if EMA_USE_ASYNC
  // 2 x (32 lanes x 16B) async copies; tracked on ASYNCcnt.
  __builtin_amdgcn_global_load_async_to_lds_b128(
      (GLOBAL_AS v4i_t*)(g + lane * 4),
      (LDS_AS v4i_t*)(buf + lane * 4), 0, 0);
  __builtin_amdgcn_global_load_async_to_lds_b128(
      (GLOBAL_AS v4i_t*)(g + 128 + lane * 4),
      (LDS_AS v4i_t*)(buf + 128 + lane * 4), 0, 0);
#else
  const float4* gs = (const float4*)g;
  float4* ls = (float4*)buf;
#pragma unroll
  for (int i = 0; i < 2; ++i) ls[lane + 32 * i] = gs[lane + 32 * i];
#endif
}

__global__ void __launch_bounds__(32) ema_main_kernel(
    const float* __restrict__ xT, const float* __restrict__ cf,
    const float* __restrict__ omega, float* __restrict__ out) {
  const int bid  = blockIdx.x;          // b*D + d
  const int b    = bid / D;
  const int d    = bid % D;
  const int lane = threadIdx.x;         // 0..31
  const int row  = lane & 15;
  const int kh   = lane >> 4;           // which half of the wave

  __shared__ float ybuf[L];             // full series accumulator (16 KB)
  __shared__ __align__(16) float xsb[2][256];   // double-buffered x staging
  __shared__ float us[256], hs[256], ys[256];

  const float* series = xT + (size_t)(b * D + d) * L;

  for (int dir = 0; dir < 2; ++dir) {
    const float* base = cf + (size_t)(dir * D + d) * CF_STRIDE;

    // A-operand register tiles (ISA 16x4 f32 layout: lanes 0-15 K=0,1;
    // lanes 16-31 K=2,3), sliced K=16 -> 4 x K=4.
    v2f tA[4], wA[4], aA[4];
#pragma unroll
    for (int kk = 0; kk < 4; ++kk) {
      const int c0 = kk * 4 + kh * 2;
      v2f t0; t0[0] = base[row * 16 + c0];       t0[1] = base[row * 16 + c0 + 1];
      v2f w0; w0[0] = base[256 + row * 16 + c0]; w0[1] = base[256 + row * 16 + c0 + 1];
      v2f a0; a0[0] = base[512 + row * 16 + c0]; a0[1] = base[512 + row * 16 + c0 + 1];
      tA[kk] = t0; wA[kk] = w0; aA[kk] = a0;
    }
    const float qC = (lane < N) ? base[768 + lane] : 0.0f;
    float carry = 0.0f;

    // Superchunk sc covers forward time [sc*256, sc*256+256) when dir==0,
    // and forward time [L-256*(sc+1), L-256*sc) (consumed reversed) when
    // dir==1. Staging is always in forward memory order (async-B128 safe).
    stage_superchunk(series + (dir == 0 ? 0 : L - 256), xsb[0], lane);

    for (int sc = 0; sc < 16; ++sc) {
      if (sc < 15) {
        const int noff = (dir == 0) ? (sc + 1) * 256 : L - 256 * (sc + 2);
        stage_superchunk(series + noff, xsb[(sc + 1) & 1], lane);
      }
#if EMA_USE_ASYNC
      if (sc < 15) __builtin_amdgcn_s_wait_asynccnt(2);  // keep next 2 in flight
      else         __builtin_amdgcn_s_wait_asynccnt(0);
#endif
      __syncthreads();
      const float* xs = xsb[sc & 1];

      // B-operand slices of X (sigma x chunk); for dir==1 consume reversed:
      // rev[j] = xs[255-j].
      v2f xB[4];
#pragma unroll
      for (int kk = 0; kk < 4; ++kk) {
        const int s0 = kk * 4 + kh * 2;          // K rows {s0, s0+1}
        const int i0 = row * 16 + s0;            // row == chunk (N index)
        v2f xv;
        if (dir == 0) { xv[0] = xs[i0];       xv[1] = xs[i0 + 1]; }
        else          { xv[0] = xs[255 - i0]; xv[1] = xs[254 - i0]; }
        xB[kk] = xv;
      }

      // U = A @ X : end-of-chunk state injection per (state n, chunk c)
      v8f U = {0.f, 0.f, 0.f, 0.f, 0.f, 0.f, 0.f, 0.f};
#pragma unroll
      for (int kk = 0; kk < 4; ++kk) U = wmma4(aA[kk], xB[kk], U);
#pragma unroll
      for (int v = 0; v < 8; ++v) us[(v + 8 * kh) * 16 + row] = U[v];
      __syncthreads();

      // Sequential state recurrence across the 16 chunks (lane n = state n).
      if (lane < N) {
        float h = carry;
#pragma unroll
        for (int c = 0; c < 16; ++c) {
          hs[c * 16 + lane] = h;                 // H_in[n][c], chunk-major
          h = qC * h + us[lane * 16 + c];
        }
        carry = h;
      }
      __syncthreads();

      // Y = T @ X + Wm @ H_in
      v8f Y = {0.f, 0.f, 0.f, 0.f, 0.f, 0.f, 0.f, 0.f};
#pragma unroll
      for (int kk = 0; kk < 4; ++kk) Y = wmma4(tA[kk], xB[kk], Y);
      v2f hB[4];
#pragma unroll
      for (int kk = 0; kk < 4; ++kk) {
        const int s0 = kk * 4 + kh * 2;
        v2f hv; hv[0] = hs[row * 16 + s0]; hv[1] = hs[row * 16 + s0 + 1];
        hB[kk] = hv;
      }
#pragma unroll
      for (int kk = 0; kk < 4; ++kk) Y = wmma4(wA[kk], hB[kk], Y);

      // Scatter Y (C/D layout: vgpr v -> M=v+8*kh, lane%16 -> N=chunk)
#pragma unroll
      for (int v = 0; v < 8; ++v) ys[row * 16 + (v + 8 * kh)] = Y[v];
      __syncthreads();

      if (dir == 0) {
        for (int i = lane; i < 256; i += 32) ybuf[sc * 256 + i] = ys[i];
      } else {
        const int ob = L - 1 - sc * 256;
        for (int i = lane; i < 256; i += 32) ybuf[ob - i] += ys[i];
      }
      __syncthreads();
    }
  }

  // Fused final write: out = fwd + bwd + omega[d]*x
  const float om = omega[d];
  for (int t = lane; t < L; t += 32)
    out[(size_t)t * (B * D) + b * D + d] = ybuf[t] + om * series[t];
}

// ---------------------------------------------------------------------------
extern "C" void kernel_launch(void* const* d_in, const int* in_sizes, int n_in,
                              void* d_out, int out_size, void* d_ws, size_t ws_size,
                              hipStream_t stream) {
  const float* x     = (const float*)d_in[0];
  const float* delta = (const float*)d_in[1];
  const float* alpha = (const float*)d_in[2];
  const float* beta  = (const float*)d_in[3];
  const float* gamma = (const float*)d_in[4];
  const float* omega = (const float*)d_in[5];
  float* out = (float*)d_out;

  float* cf = (float*)d_ws;                                  // 6.4 MB
  float* xT = (float*)((char*)d_ws + CF_BYTES);              // 64 MB

  ema_coef_kernel<<<KD, 256, 0, stream>>>(delta, alpha, beta, gamma, cf);
  ema_transpose_kernel<<<dim3(L / 32, D / 32, B), dim3(32, 8), 0, stream>>>(x, xT);
  ema_main_kernel<<<B * D, 32, 0, stream>>>(xT, cf, omega, out);
}